// MultiHeadAttention_45466523795869
// MI455X (gfx1250) — compile-verified
//
#include <hip/hip_runtime.h>
#include <hip/hip_bf16.h>
#include <stdint.h>

#define B_ 8
#define L_ 1024
#define D_ 1024
#define H_ 16
#define DK_ 64
#define NEG_ (-1e10f)

typedef float v2f __attribute__((ext_vector_type(2)));
typedef float v8f __attribute__((ext_vector_type(8)));

__device__ __forceinline__ v8f wmma_f32(v2f a, v2f b, v8f c) {
  // D = A(16x4) * B(4x16) + C(16x16), all f32, wave32
  return __builtin_amdgcn_wmma_f32_16x16x4_f32(
      false, a, false, b, (short)0, c, false, false);
}

// Async DMA: global -> LDS, 16B per lane, tracked by ASYNCcnt.
__device__ __forceinline__ void async_b128(void* lds, const void* gmem) {
  const unsigned la = (unsigned)(uintptr_t)lds;  // low 32 bits = LDS byte offset
  asm volatile("global_load_async_to_lds_b128 %0, %1, off"
               :: "v"(la), "v"(gmem) : "memory");
}
// Partial ASYNCcnt wait: allow `n` newest loads to remain in flight.
#define WAIT_ASYNC(n) asm volatile("s_wait_asynccnt " #n ::: "memory")

// Padded LDS strides: stride % 64 chosen so 16 lanes (rows) hit distinct banks.
#define ALD 68     // 68 % 64 = 4 -> conflict-free for 16-lane row access
#define SLD 1044   // 1044 % 64 = 20 -> m*20 mod 64 all distinct

// ---------------------------------------------------------------------------
// Generic GEMM: out = X[M x 1024] @ W^T (B[k][n] = W[n][k]).
// Block tile 64x128, 8 waves (m-tile x n-group); K in 64-wide tiles,
// prefetch-depth-2 async DMA double buffering + register-level ds pipelining.
// mode 0: out row-major [M,1024];  mode 1: head scatter to [B,H,L,64].
// grid (M/64, 1024/128), block 256.
// ---------------------------------------------------------------------------
__global__ __launch_bounds__(256) void gemm_kernel(
    const float* __restrict__ X, const float* __restrict__ W,
    float* __restrict__ out, int mode) {
  extern __shared__ float smem[];
  float* bufA = smem;                  // [2][64 * ALD]
  float* bufB = smem + 2 * 64 * ALD;   // [2][128 * ALD]

  const int row0 = blockIdx.x * 64;
  const int n0 = blockIdx.y * 128;
  const int tid = threadIdx.x;
  const int lane = tid & 31, wave = tid >> 5;
  const int half = lane >> 4, l16 = lane & 15;
  const int mt = wave >> 1, ng = wave & 1;
  const int sr = tid >> 2, sc4 = (tid & 3) * 4;  // staging row / float4-col

  auto stage = [&](int t, int buf) {  // 3 async loads per thread
    const int kb0 = t * 64;
    async_b128(&bufA[buf * 64 * ALD + sr * ALD + sc4],
               X + (size_t)(row0 + sr) * D_ + kb0 + sc4);
    async_b128(&bufB[buf * 128 * ALD + sr * ALD + sc4],
               W + (size_t)(n0 + sr) * D_ + kb0 + sc4);
    async_b128(&bufB[buf * 128 * ALD + (sr + 64) * ALD + sc4],
               W + (size_t)(n0 + sr + 64) * D_ + kb0 + sc4);
  };

  v8f acc[4] = {{}, {}, {}, {}};
  stage(0, 0);
  stage(1, 1);

  for (int t = 0; t < 16; ++t) {
    const int buf = t & 1;
    // tile t must have landed; tile t+1 (3 loads) may stay in flight.
    if (t + 1 < 16) { WAIT_ASYNC(3); } else { WAIT_ASYNC(0); }
    __syncthreads();
    const float* tA = &bufA[buf * 64 * ALD + (mt * 16 + l16) * ALD];
    const float* tB = &bufB[buf * 128 * ALD];

    // Register double-buffered fragments: ds loads run one k-step ahead.
    v2f a_c, b_c[4];
    {
      const int kk = 2 * half;
      a_c.x = tA[kk]; a_c.y = tA[kk + 1];
#pragma unroll
      for (int u = 0; u < 4; ++u) {
        const int nl = (ng * 4 + u) * 16 + l16;
        b_c[u].x = tB[nl * ALD + kk]; b_c[u].y = tB[nl * ALD + kk + 1];
      }
    }
#pragma unroll
    for (int k0 = 0; k0 < 64; k0 += 4) {
      v2f a_n, b_n[4];
      if (k0 + 4 < 64) {
        const int kk = k0 + 4 + 2 * half;
        a_n.x = tA[kk]; a_n.y = tA[kk + 1];
#pragma unroll
        for (int u = 0; u < 4; ++u) {
          const int nl = (ng * 4 + u) * 16 + l16;
          b_n[u].x = tB[nl * ALD + kk]; b_n[u].y = tB[nl * ALD + kk + 1];
        }
      }
#pragma unroll
      for (int u = 0; u < 4; ++u) acc[u] = wmma_f32(a_c, b_c[u], acc[u]);
      if (k0 + 4 < 64) {
        a_c = a_n;
#pragma unroll
        for (int u = 0; u < 4; ++u) b_c[u] = b_n[u];
      }
    }
    __syncthreads();
    if (t + 2 < 16) stage(t + 2, buf);  // refill the buffer just freed
  }

#pragma unroll
  for (int u = 0; u < 4; ++u) {
    const int n = n0 + (ng * 4 + u) * 16 + l16;
    for (int r = 0; r < 8; ++r) {
      const int grow = row0 + mt * 16 + r + 8 * half;
      if (mode == 0) {
        out[(size_t)grow * D_ + n] = acc[u][r];
      } else {
        const int bi = grow >> 10, l = grow & 1023;
        const int h = n >> 6, d = n & 63;
        out[(((size_t)(bi * H_ + h)) * L_ + l) * DK_ + d] = acc[u][r];
      }
    }
  }
}

// ---------------------------------------------------------------------------
// Attention: scores (K-tiles async, depth-2) -> mask -> softmax -> NT attn
// writeout -> ctx = P @ V with 8-wave K-split + LDS reduction.
// grid (L/16, H, B), block 256.
// ---------------------------------------------------------------------------
__global__ __launch_bounds__(256) void attn_kernel(
    const float* __restrict__ Qh, const float* __restrict__ Kh,
    const float* __restrict__ Vh, const unsigned char* __restrict__ mask,
    float* __restrict__ attn_out, float* __restrict__ ctx) {
  extern __shared__ float smem[];
  float* ldsS = smem;                         // 16 x SLD score strip
  float* ldsQ = smem + 16 * SLD;              // 16 x ALD (later: ctx partials)
  float* ldsKV = smem + 16 * SLD + 16 * ALD;  // 4 slots of [64 * ALD]
  __shared__ float red[256];
  __shared__ float rowstat[16];

  const int q0 = blockIdx.x * 16;
  const int h = blockIdx.y, b = blockIdx.z;
  const size_t base = ((size_t)(b * H_ + h)) * L_ * DK_;

  const int tid = threadIdx.x;
  const int lane = tid & 31, wave = tid >> 5;
  const int half = lane >> 4, l16 = lane & 15;
  const int sr = tid >> 2, sc4 = (tid & 3) * 4;

  // Q strip [16 x 64] + first two K tiles via async DMA.
  {
    const int qr = tid >> 4, qc4 = (tid & 15) * 4;
    async_b128(&ldsQ[qr * ALD + qc4],
               Qh + base + (size_t)(q0 + qr) * DK_ + qc4);
  }
  auto stageK = [&](int j, int buf) {  // 2 async loads per thread
    const int key0 = j * 128;
    async_b128(&ldsKV[(buf * 2 + 0) * 64 * ALD + sr * ALD + sc4],
               Kh + base + (size_t)(key0 + sr) * DK_ + sc4);
    async_b128(&ldsKV[(buf * 2 + 1) * 64 * ALD + sr * ALD + sc4],
               Kh + base + (size_t)(key0 + 64 + sr) * DK_ + sc4);
  };
  stageK(0, 0);
  stageK(1, 1);

  // Scores: per j, 8 waves cover 128 keys (wave -> 16 keys).
  for (int j = 0; j < 8; ++j) {
    const int buf = j & 1;
    if (j + 1 < 8) { WAIT_ASYNC(2); } else { WAIT_ASYNC(0); }
    __syncthreads();
    const float* tK = &ldsKV[buf * 128 * ALD + wave * 16 * ALD];
    v8f acc = {};
    v2f a_c, b_c;
    {
      const int kk = 2 * half;
      a_c.x = ldsQ[l16 * ALD + kk]; a_c.y = ldsQ[l16 * ALD + kk + 1];
      b_c.x = tK[l16 * ALD + kk];   b_c.y = tK[l16 * ALD + kk + 1];
    }
#pragma unroll
    for (int k0 = 0; k0 < DK_; k0 += 4) {
      v2f a_n, b_n;
      if (k0 + 4 < DK_) {
        const int kk = k0 + 4 + 2 * half;
        a_n.x = ldsQ[l16 * ALD + kk]; a_n.y = ldsQ[l16 * ALD + kk + 1];
        b_n.x = tK[l16 * ALD + kk];   b_n.y = tK[l16 * ALD + kk + 1];
      }
      acc = wmma_f32(a_c, b_c, acc);
      if (k0 + 4 < DK_) { a_c = a_n; b_c = b_n; }
    }
    const int kc = j * 128 + wave * 16 + l16;
    for (int r = 0; r < 8; ++r) {
      const int lrow = r + 8 * half;
      float s = acc[r] * 0.125f;
      const unsigned char mbit = __builtin_nontemporal_load(
          mask + ((size_t)b * L_ + (q0 + lrow)) * L_ + kc);
      if (mbit) s = NEG_;
      ldsS[lrow * SLD + kc] = s;
    }
    __syncthreads();
    if (j + 2 < 8) stageK(j + 2, buf);
  }
  __syncthreads();

  // Row softmax over 1024 cols: 16 threads per row.
  const int r = tid >> 4, t = tid & 15;
  float m = -3.4e38f;
  for (int c = t; c < L_; c += 16) m = fmaxf(m, ldsS[r * SLD + c]);
  red[r * 16 + t] = m;
  __syncthreads();
  if (t == 0) {
    float mm = red[r * 16];
    for (int i = 1; i < 16; ++i) mm = fmaxf(mm, red[r * 16 + i]);
    rowstat[r] = mm;
  }
  __syncthreads();
  const float rowmax = rowstat[r];
  float lsum = 0.f;
  for (int c = t; c < L_; c += 16) {
    const float e = __expf(ldsS[r * SLD + c] - rowmax);
    ldsS[r * SLD + c] = e;
    lsum += e;
  }
  red[r * 16 + t] = lsum;
  __syncthreads();
  if (t == 0) {
    float s = 0.f;
    for (int i = 0; i < 16; ++i) s += red[r * 16 + i];
    rowstat[r] = 1.0f / s;
  }
  __syncthreads();
  const float inv = rowstat[r];
  float* arow = attn_out + ((size_t)(h * B_ + b) * L_ + (q0 + r)) * L_;
  for (int c = t; c < L_; c += 16) {
    const float p = ldsS[r * SLD + c] * inv;
    ldsS[r * SLD + c] = p;
    __builtin_nontemporal_store(p, arow + c);  // 537MB stream: NT, bypass cache
  }
  __syncthreads();

  // ctx[16 x 64] = P[16 x 1024] @ V[1024 x 64].
  // 8 waves: group g = wave>>2 takes K-half g*512..; two V tiles staged/iter.
  auto stageVpair = [&](int kb, int buf) {  // 2 async loads per thread
    async_b128(&ldsKV[(buf * 2 + 0) * 64 * ALD + sr * ALD + sc4],
               Vh + base + (size_t)(kb * 64 + sr) * DK_ + sc4);
    async_b128(&ldsKV[(buf * 2 + 1) * 64 * ALD + sr * ALD + sc4],
               Vh + base + (size_t)(512 + kb * 64 + sr) * DK_ + sc4);
  };
  stageVpair(0, 0);
  stageVpair(1, 1);

  v8f vacc = {};
  const int g = wave >> 2;
  const int col0 = (wave & 3) * 16;
  for (int kb = 0; kb < 8; ++kb) {
    const int buf = kb & 1;
    if (kb + 1 < 8) { WAIT_ASYNC(2); } else { WAIT_ASYNC(0); }
    __syncthreads();
    const float* tV = &ldsKV[(buf * 2 + g) * 64 * ALD];
    const int kbase = (g * 8 + kb) * 64;
    v2f a_c, b_c;
    {
      const int kk = 2 * half;
      a_c.x = ldsS[l16 * SLD + kbase + kk];
      a_c.y = ldsS[l16 * SLD + kbase + kk + 1];
      b_c.x = tV[kk * ALD + col0 + l16];
      b_c.y = tV[(kk + 1) * ALD + col0 + l16];
    }
#pragma unroll
    for (int k0 = 0; k0 < 64; k0 += 4) {
      v2f a_n, b_n;
      if (k0 + 4 < 64) {
        const int kk = k0 + 4 + 2 * half;
        a_n.x = ldsS[l16 * SLD + kbase + kk];
        a_n.y = ldsS[l16 * SLD + kbase + kk + 1];
        b_n.x = tV[kk * ALD + col0 + l16];
        b_n.y = tV[(kk + 1) * ALD + col0 + l16];
      }
      vacc = wmma_f32(a_c, b_c, vacc);
      if (k0 + 4 < 64) { a_c = a_n; b_c = b_n; }
    }
    __syncthreads();
    if (kb + 2 < 8) stageVpair(kb + 2, buf);
  }

  // Reduce the two K-halves through LDS (reuse dead ldsQ area), then store.
  if (g == 1) {
    for (int rr = 0; rr < 8; ++rr)
      ldsQ[(rr + 8 * half) * ALD + col0 + l16] = vacc[rr];
  }
  __syncthreads();
  if (g == 0) {
    for (int rr = 0; rr < 8; ++rr) {
      const int lrow = rr + 8 * half;
      ctx[((size_t)b * L_ + q0 + lrow) * (H_ * DK_) + h * DK_ + col0 + l16] =
          vacc[rr] + ldsQ[lrow * ALD + col0 + l16];
    }
  }
}

// ---------------------------------------------------------------------------
// LayerNorm epilogue: y = LN(fcout + fc_b + resid).  grid 8192, block 256.
// ---------------------------------------------------------------------------
__global__ __launch_bounds__(256) void ln_kernel(
    const float* __restrict__ fcout, const float* __restrict__ resid,
    const float* __restrict__ fc_b, const float* __restrict__ ln_g,
    const float* __restrict__ ln_b, float* __restrict__ y) {
  __shared__ float red[256];
  const int row = blockIdx.x;
  const int tid = threadIdx.x;
  const float* fp = fcout + (size_t)row * D_;
  const float* rp = resid + (size_t)row * D_;

  float xv[4];
  float lsum = 0.f;
#pragma unroll
  for (int i = 0; i < 4; ++i) {
    const int c = tid + i * 256;
    xv[i] = fp[c] + fc_b[c] + rp[c];
    lsum += xv[i];
  }
  red[tid] = lsum;
  __syncthreads();
  for (int s = 128; s > 0; s >>= 1) {
    if (tid < s) red[tid] += red[tid + s];
    __syncthreads();
  }
  const float mu = red[0] * (1.0f / D_);
  __syncthreads();
  float lvar = 0.f;
#pragma unroll
  for (int i = 0; i < 4; ++i) {
    const float d = xv[i] - mu;
    lvar += d * d;
  }
  red[tid] = lvar;
  __syncthreads();
  for (int s = 128; s > 0; s >>= 1) {
    if (tid < s) red[tid] += red[tid + s];
    __syncthreads();
  }
  const float rstd = rsqrtf(red[0] * (1.0f / D_) + 1e-5f);
#pragma unroll
  for (int i = 0; i < 4; ++i) {
    const int c = tid + i * 256;
    y[(size_t)row * D_ + c] = (xv[i] - mu) * rstd * ln_g[c] + ln_b[c];
  }
}

// ---------------------------------------------------------------------------
extern "C" void kernel_launch(void* const* d_in, const int* in_sizes, int n_in,
                              void* d_out, int out_size, void* d_ws,
                              size_t ws_size, hipStream_t stream) {
  const float* q = (const float*)d_in[0];
  const float* k = (const float*)d_in[1];
  const float* v = (const float*)d_in[2];
  const unsigned char* mask = (const unsigned char*)d_in[3];  // jax bool = 1B
  const float* w_qs = (const float*)d_in[4];
  const float* w_ks = (const float*)d_in[5];
  const float* w_vs = (const float*)d_in[6];
  const float* fc_w = (const float*)d_in[7];
  const float* fc_b = (const float*)d_in[8];
  const float* ln_g = (const float*)d_in[9];
  const float* ln_b = (const float*)d_in[10];

  const size_t NTOK = (size_t)B_ * L_;  // 8192 rows
  const size_t NEL = NTOK * D_;         // 8M floats
  float* ws = (float*)d_ws;
  float* Qh = ws;                       // [B,H,L,64]
  float* Kh = ws + NEL;
  float* Vh = ws + 2 * NEL;
  float* ctx = ws + 3 * NEL;            // [B*L, H*64]
  float* fcout = ws + 4 * NEL;          // [B*L, D]

  float* y = (float*)d_out;             // [B,L,D]
  float* attn = y + NEL;                // [H*B, L, L]

  const size_t smem_g =
      (size_t)(2 * 64 * ALD + 2 * 128 * ALD) * sizeof(float);  // ~102KB
  const dim3 ggrid(NTOK / 64, D_ / 128);
  gemm_kernel<<<ggrid, 256, smem_g, stream>>>(q, w_qs, Qh, 1);
  gemm_kernel<<<ggrid, 256, smem_g, stream>>>(k, w_ks, Kh, 1);
  gemm_kernel<<<ggrid, 256, smem_g, stream>>>(v, w_vs, Vh, 1);

  const size_t smem_a =
      (size_t)(16 * SLD + 16 * ALD + 4 * 64 * ALD) * sizeof(float);  // ~137KB
  attn_kernel<<<dim3(L_ / 16, H_, B_), 256, smem_a, stream>>>(
      Qh, Kh, Vh, mask, attn, ctx);

  gemm_kernel<<<ggrid, 256, smem_g, stream>>>(ctx, fc_w, fcout, 0);

  ln_kernel<<<dim3(NTOK), 256, 0, stream>>>(fcout, q, fc_b, ln_g, ln_b, y);
}